// PTRoadGraphEmbedding_21199958573627
// MI455X (gfx1250) — compile-verified
//
#include <hip/hip_runtime.h>
#include <math.h>

typedef __attribute__((ext_vector_type(2))) float v2f;
typedef __attribute__((ext_vector_type(8))) float v8f;

#define N_NODES 50000
#define N_PAD   50176          // multiple of 256 (block covers 256 rows)
#define N_EDGES 800000
#define FEAT    256
#define HEADS   8
#define HDIM    32
#define NEG_ATTN 0.2f
#define NEG_ACT  0.01f

#define KSTAGE   128           // K rows staged in LDS at a time (64 k-pairs)
#define PSTRIDE  80            // float2 stride per k-pair row: 64 data + 16 pad
                               // 160 floats => p -> p+1 shifts banks by 32 (conflict-free)

// -------- order-independent float atomic max (sign-aware int/uint trick) ----
__device__ __forceinline__ void atomicMaxFloat(float* addr, float val) {
    if (val >= 0.0f) {
        atomicMax((int*)addr, __float_as_int(val));
    } else {
        atomicMin((unsigned int*)addr, __float_as_uint(val));
    }
}

// ---------------- GEMM: h = x @ W via V_WMMA_F32_16X16X4_F32 ----------------
// Block = 8 waves, computes 256 rows x 64 cols. W panel staged in LDS in
// k-pair-interleaved form: lds2[p*PSTRIDE + n] = {W[2p][n], W[2p+1][n]}.
// Each wave computes 32x64 (2 row strips share the B fragments); each B
// fragment is then a single aligned ds_load_b64.
__global__ void gat_gemm_wmma(const float* __restrict__ x,
                              const float* __restrict__ W,
                              float* __restrict__ h)
{
    __shared__ v2f lds2[(KSTAGE / 2) * PSTRIDE];   // 40 KB

    const int group = blockIdx.x >> 2;           // row group of 256
    const int ct    = blockIdx.x & 3;            // column tile of 64
    const int n0    = ct * 64;
    const int w     = threadIdx.x >> 5;          // wave id 0..7
    const int lane  = threadIdx.x & 31;
    const int half  = lane >> 4;                 // 0: K+0..1, 1: K+2..3
    const int l     = lane & 15;

    const int m0 = group * 256 + w * 32;         // this wave: rows m0..m0+31

    int arow0 = m0 + l;        if (arow0 >= N_NODES) arow0 = N_NODES - 1;
    int arow1 = m0 + 16 + l;   if (arow1 >= N_NODES) arow1 = N_NODES - 1;
    const float* __restrict__ ap0 = x + (size_t)arow0 * FEAT + 2 * half;
    const float* __restrict__ ap1 = x + (size_t)arow1 * FEAT + 2 * half;

    v8f c00 = {}; v8f c01 = {}; v8f c02 = {}; v8f c03 = {};
    v8f c10 = {}; v8f c11 = {}; v8f c12 = {}; v8f c13 = {};

    for (int ks = 0; ks < FEAT; ks += KSTAGE) {
        __syncthreads();   // protect LDS WAR vs previous stage's compute
        // cooperative stage: 1024 tasks = 64 pairs x 16 float4-columns
#pragma unroll
        for (int i = 0; i < 4; ++i) {
            int task = i * 256 + threadIdx.x;         // 0..1023
            int p    = task >> 4;                     // k-pair 0..63
            int n    = (task & 15) * 4;               // col 0..60
            const float* __restrict__ wr = W + (size_t)(ks + 2 * p) * FEAT + n0 + n;
            float4 wa = *(const float4*)(wr);
            float4 wb = *(const float4*)(wr + FEAT);
            v2f* __restrict__ dp = &lds2[p * PSTRIDE + n];
            v2f q0; q0.x = wa.x; q0.y = wb.x; dp[0] = q0;
            v2f q1; q1.x = wa.y; q1.y = wb.y; dp[1] = q1;
            v2f q2; q2.x = wa.z; q2.y = wb.z; dp[2] = q2;
            v2f q3; q3.x = wa.w; q3.y = wb.w; dp[3] = q3;
        }
        __syncthreads();

        for (int kb = 0; kb < KSTAGE; kb += 4) {
            v2f a0 = *(const v2f*)(ap0 + ks + kb);
            v2f a1 = *(const v2f*)(ap1 + ks + kb);

            // this lane's pair row: k = kb + 2*half  ->  p = kb/2 + half
            const v2f* __restrict__ bp = &lds2[((kb >> 1) + half) * PSTRIDE + l];
            v2f b0 = bp[0];
            v2f b1 = bp[16];
            v2f b2 = bp[32];
            v2f b3 = bp[48];

            c00 = __builtin_amdgcn_wmma_f32_16x16x4_f32(false, a0, false, b0, (short)0, c00, false, false);
            c01 = __builtin_amdgcn_wmma_f32_16x16x4_f32(false, a0, false, b1, (short)0, c01, false, false);
            c02 = __builtin_amdgcn_wmma_f32_16x16x4_f32(false, a0, false, b2, (short)0, c02, false, false);
            c03 = __builtin_amdgcn_wmma_f32_16x16x4_f32(false, a0, false, b3, (short)0, c03, false, false);
            c10 = __builtin_amdgcn_wmma_f32_16x16x4_f32(false, a1, false, b0, (short)0, c10, false, false);
            c11 = __builtin_amdgcn_wmma_f32_16x16x4_f32(false, a1, false, b1, (short)0, c11, false, false);
            c12 = __builtin_amdgcn_wmma_f32_16x16x4_f32(false, a1, false, b2, (short)0, c12, false, false);
            c13 = __builtin_amdgcn_wmma_f32_16x16x4_f32(false, a1, false, b3, (short)0, c13, false, false);
        }
    }

    // C/D layout: VGPR r -> row (+8*half + r), col n0 + j*16 + l
    float* __restrict__ hp0 = h + (size_t)(m0 + half * 8) * FEAT + n0 + l;
    float* __restrict__ hp1 = hp0 + (size_t)16 * FEAT;
#pragma unroll
    for (int r = 0; r < 8; ++r) {
        hp0[r * FEAT + 0]  = c00[r];
        hp0[r * FEAT + 16] = c01[r];
        hp0[r * FEAT + 32] = c02[r];
        hp0[r * FEAT + 48] = c03[r];
        hp1[r * FEAT + 0]  = c10[r];
        hp1[r * FEAT + 16] = c11[r];
        hp1[r * FEAT + 32] = c12[r];
        hp1[r * FEAT + 48] = c13[r];
    }
}

// ---------------- per-node attention projections el/er ----------------------
__global__ void gat_attn_coeff(const float* __restrict__ h,
                               const float* __restrict__ al,
                               const float* __restrict__ ar,
                               float* __restrict__ el,
                               float* __restrict__ er)
{
    int t = blockIdx.x * blockDim.x + threadIdx.x;
    if (t >= N_NODES * HEADS) return;
    int node = t >> 3, hd = t & 7;
    const float4* __restrict__ hp  = (const float4*)(h + (size_t)node * FEAT + hd * HDIM);
    const float4* __restrict__ alp = (const float4*)(al + hd * HDIM);
    const float4* __restrict__ arp = (const float4*)(ar + hd * HDIM);
    float sl = 0.f, sr = 0.f;
#pragma unroll
    for (int i = 0; i < HDIM / 4; ++i) {
        float4 v = hp[i];
        float4 a = alp[i];
        float4 r = arp[i];
        sl += v.x * a.x + v.y * a.y + v.z * a.z + v.w * a.w;
        sr += v.x * r.x + v.y * r.y + v.z * r.z + v.w * r.w;
    }
    el[t] = sl;
    er[t] = sr;
}

// ---------------- zero accumulator, init segment max/sum --------------------
__global__ void gat_init(float4* __restrict__ acc4,
                         float* __restrict__ m,
                         float* __restrict__ s)
{
    int t = blockIdx.x * blockDim.x + threadIdx.x;
    if (t < N_NODES * (FEAT / 4)) {
        float4 z; z.x = 0.f; z.y = 0.f; z.z = 0.f; z.w = 0.f;
        acc4[t] = z;
    }
    if (t < N_NODES * HEADS) {
        m[t] = -__builtin_inff();
        s[t] = 0.0f;
    }
}

// ---------------- edge logits + segment max ---------------------------------
__global__ void gat_edge_logits(const int* __restrict__ src,
                                const int* __restrict__ dst,
                                const float* __restrict__ el,
                                const float* __restrict__ er,
                                float* __restrict__ ebuf,
                                float* __restrict__ m)
{
    int t = blockIdx.x * blockDim.x + threadIdx.x;
    if (t >= N_EDGES * HEADS) return;
    int e = t >> 3, hd = t & 7;
    int si = src[e], di = dst[e];
    float v = el[si * HEADS + hd] + er[di * HEADS + hd];
    v = (v > 0.0f) ? v : NEG_ATTN * v;           // leaky_relu(0.2)
    ebuf[t] = v;
    atomicMaxFloat(&m[di * HEADS + hd], v);
}

// ---------------- exp(e - m[dst]) + segment sum -----------------------------
__global__ void gat_edge_exp(const int* __restrict__ dst,
                             const float* __restrict__ m,
                             float* __restrict__ ebuf,
                             float* __restrict__ s)
{
    int t = blockIdx.x * blockDim.x + threadIdx.x;
    if (t >= N_EDGES * HEADS) return;
    int e = t >> 3, hd = t & 7;
    int di = dst[e];
    float ex = __expf(ebuf[t] - m[di * HEADS + hd]);
    ebuf[t] = ex;
    atomicAdd(&s[di * HEADS + hd], ex);
}

// ---------------- weighted message aggregation (4 feats / thread) -----------
__global__ void gat_aggregate(const int* __restrict__ src,
                              const int* __restrict__ dst,
                              const float* __restrict__ h,
                              const float* __restrict__ ebuf,
                              const float* __restrict__ s,
                              float* __restrict__ out)
{
    long long t = (long long)blockIdx.x * blockDim.x + threadIdx.x;
    if (t >= (long long)N_EDGES * (FEAT / 4)) return;
    int e  = (int)(t >> 6);
    int q  = (int)(t & 63);
    int f  = q * 4;
    int hd = q >> 3;
    int si = src[e], di = dst[e];
    float a = ebuf[e * HEADS + hd] / s[di * HEADS + hd];
    float4 hv = *(const float4*)(h + (size_t)si * FEAT + f);
    float* __restrict__ op = out + (size_t)di * FEAT + f;
    atomicAdd(op + 0, hv.x * a);
    atomicAdd(op + 1, hv.y * a);
    atomicAdd(op + 2, hv.z * a);
    atomicAdd(op + 3, hv.w * a);
}

// ---------------- bias + optional inter-layer leaky_relu --------------------
__global__ void gat_bias_act(float4* __restrict__ out4,
                             const float4* __restrict__ b4,
                             int apply_act)
{
    int t = blockIdx.x * blockDim.x + threadIdx.x;
    if (t >= N_NODES * (FEAT / 4)) return;
    int q = t & 63;
    float4 v = out4[t];
    float4 bb = b4[q];
    v.x += bb.x; v.y += bb.y; v.z += bb.z; v.w += bb.w;
    if (apply_act) {
        v.x = (v.x > 0.f) ? v.x : NEG_ACT * v.x;
        v.y = (v.y > 0.f) ? v.y : NEG_ACT * v.y;
        v.z = (v.z > 0.f) ? v.z : NEG_ACT * v.z;
        v.w = (v.w > 0.f) ? v.w : NEG_ACT * v.w;
    }
    out4[t] = v;
}

extern "C" void kernel_launch(void* const* d_in, const int* in_sizes, int n_in,
                              void* d_out, int out_size, void* d_ws, size_t ws_size,
                              hipStream_t stream)
{
    (void)in_sizes; (void)n_in; (void)out_size; (void)ws_size;
    const float* x   = (const float*)d_in[0];
    const int*   src = (const int*)d_in[1];
    const int*   dst = (const int*)d_in[2];
    const float* W   = (const float*)d_in[3];
    const float* al  = (const float*)d_in[4];
    const float* ar  = (const float*)d_in[5];
    const float* b   = (const float*)d_in[6];
    float* out = (float*)d_out;

    // workspace carve-up (~135 MB)
    float* h    = (float*)d_ws;                       // N_PAD  * FEAT
    float* bufA = h    + (size_t)N_PAD  * FEAT;       // N_NODES* FEAT
    float* elb  = bufA + (size_t)N_NODES * FEAT;      // N_NODES* HEADS
    float* erb  = elb  + (size_t)N_NODES * HEADS;
    float* mb   = erb  + (size_t)N_NODES * HEADS;
    float* sb   = mb   + (size_t)N_NODES * HEADS;
    float* ebuf = sb   + (size_t)N_NODES * HEADS;     // N_EDGES* HEADS

    const int TB = 256;
    const int gemm_blocks = (N_PAD / 256) * 4;                 // 196 groups x 4 col tiles
    const int nh_blocks   = (N_NODES * HEADS + TB - 1) / TB;
    const int eh_blocks   = (N_EDGES * HEADS + TB - 1) / TB;
    const int nf4_blocks  = (N_NODES * (FEAT / 4) + TB - 1) / TB;
    const long long aggT  = (long long)N_EDGES * (FEAT / 4);
    const int agg_blocks  = (int)((aggT + TB - 1) / TB);

    for (int L = 0; L < 3; ++L) {
        const float* xin  = (L == 0) ? x : bufA;
        float*       yout = (L == 2) ? out : bufA;  // input buffer is free after GEMM
        int act = (L < 2);

        gat_gemm_wmma<<<gemm_blocks, TB, 0, stream>>>(xin, W + (size_t)L * FEAT * FEAT, h);
        gat_attn_coeff<<<nh_blocks, TB, 0, stream>>>(h, al + L * FEAT, ar + L * FEAT, elb, erb);
        gat_init<<<nf4_blocks, TB, 0, stream>>>((float4*)yout, mb, sb);
        gat_edge_logits<<<eh_blocks, TB, 0, stream>>>(src, dst, elb, erb, ebuf, mb);
        gat_edge_exp<<<eh_blocks, TB, 0, stream>>>(dst, mb, ebuf, sb);
        gat_aggregate<<<agg_blocks, TB, 0, stream>>>(src, dst, h, ebuf, sb, yout);
        gat_bias_act<<<nf4_blocks, TB, 0, stream>>>((float4*)yout, (const float4*)(b + L * FEAT), act);
    }
}